// LossRegressionGaussianWithCorrelations_44332652429859
// MI455X (gfx1250) — compile-verified
//
#include <hip/hip_runtime.h>
#include <math.h>

// MI455X / gfx1250 implementation.
//  - loss_lik: streaming float4 reduction over N=16.7M (HBM-bound, ~5.5us floor at 23.3 TB/s)
//  - loss_var: blocked right-looking Cholesky (NB=64) on a workspace copy of Sigma.
//    Trailing SYRK uses V_WMMA_F32_16X16X4_F32 with 2x2 register blocking:
//    each wave owns a 32x32 macro-tile (four 16x16 accumulators), so every
//    A/B fragment load feeds two WMMAs (2x L2-compute intensity vs 1-tile waves).
//    Whole 16MB matrix is L2-resident (192MB L2).
//  - forward solve + z.z + sum(log diag) in one single-workgroup kernel.
//  - deterministic: fixed-order tree reductions only, no float atomics.

typedef __attribute__((ext_vector_type(2))) float v2f;
typedef __attribute__((ext_vector_type(8))) float v8f;

#define Dn 2048
#define NB 64
#define LOG2PI 1.8378770664093453f

// ---------------- copy Sigma -> workspace (don't mutate inputs) ----------------
__global__ void copy_mat(const float* __restrict__ src, float* __restrict__ dst) {
    int i = blockIdx.x * blockDim.x + threadIdx.x;
    int stride = gridDim.x * blockDim.x;
    const float4* s4 = (const float4*)src;
    float4* d4 = (float4*)dst;
    const int n4 = Dn * Dn / 4;
    for (; i < n4; i += stride) d4[i] = s4[i];
}

// ---------------- unblocked Cholesky of a 64x64 diagonal block ----------------
__global__ void chol_diag(float* __restrict__ L, int off) {
    __shared__ float s[NB][NB + 1];
    int t = threadIdx.x; // 64 threads, thread t owns row t
    for (int c = 0; c < NB; ++c) s[t][c] = L[(size_t)(off + t) * Dn + off + c];
    __syncthreads();
    for (int k = 0; k < NB; ++k) {
        if (t == k) s[k][k] = sqrtf(s[k][k]);
        __syncthreads();
        if (t > k) s[t][k] /= s[k][k];
        __syncthreads();
        if (t > k) {
            float ltk = s[t][k];
            for (int c = k + 1; c <= t; ++c) s[t][c] -= ltk * s[c][k];
        }
        __syncthreads();
    }
    for (int c = 0; c < NB; ++c) L[(size_t)(off + t) * Dn + off + c] = s[t][c];
}

// ---------------- panel TRSM: L21 = A21 * L11^-T (row-parallel) ----------------
__global__ void trsm_panel(float* __restrict__ L, int off) {
    __shared__ float l11[NB][NB + 1];
    __shared__ float row[NB][NB + 1];
    int t = threadIdx.x;                              // 64 threads
    int r = off + NB + blockIdx.x * NB + t;           // global row this thread owns
    for (int c = 0; c < NB; ++c) {
        l11[t][c] = L[(size_t)(off + t) * Dn + off + c];
        row[t][c] = L[(size_t)r * Dn + off + c];
    }
    __syncthreads();
    for (int c = 0; c < NB; ++c) {
        float v = row[t][c];
        for (int k = 0; k < c; ++k) v -= row[t][k] * l11[c][k];
        row[t][c] = v / l11[c][c];
    }
    for (int c = 0; c < NB; ++c) L[(size_t)r * Dn + off + c] = row[t][c];
}

// ---------------- trailing SYRK update via f32 WMMA: A22 -= L21 * L21^T ----------------
// One wave per 32x32 macro-tile (2x2 register blocking of 16x16 WMMA tiles).
__global__ void syrk_wmma(float* __restrict__ L, int off, int ntri) {
    int wave = blockIdx.x * (blockDim.x >> 5) + (threadIdx.x >> 5);
    if (wave >= ntri) return;          // wave-uniform exit; EXEC stays all-1 for WMMA
    int lane = threadIdx.x & 31;

    // linear triangular index -> (ti, tj), tj <= ti (32-granularity macro-tiles)
    int ti = (int)((sqrtf(8.0f * (float)wave + 1.0f) - 1.0f) * 0.5f);
    while ((ti + 1) * (ti + 2) / 2 <= wave) ++ti;
    while (ti * (ti + 1) / 2 > wave) --ti;
    int tj = wave - ti * (ti + 1) / 2;

    int off2 = off + NB;               // trailing-matrix origin
    int gi = off2 + ti * 32;
    int gj = off2 + tj * 32;
    int m   = lane & 15;
    int klo = (lane < 16) ? 0 : 2;     // ISA 16x4 f32 A layout: lanes 0-15 hold K0/K1, 16-31 K2/K3
    int hi8 = (lane < 16) ? 0 : 8;     // C/D layout: VGPR r -> M=r (lanes 0-15) / M=r+8 (16-31)

    // load 4 accumulator tiles (documented 16x16 f32 C/D layout)
    v8f c00, c01, c10, c11;
#pragma unroll
    for (int r = 0; r < 8; ++r) {
        size_t row0 = (size_t)(gi + r + hi8) * Dn;
        size_t row1 = (size_t)(gi + 16 + r + hi8) * Dn;
        c00[r] = L[row0 + gj + m];
        c01[r] = L[row0 + gj + 16 + m];
        c10[r] = L[row1 + gj + m];
        c11[r] = L[row1 + gj + 16 + m];
    }

    const float* Arow0 = L + (size_t)(gi + m) * Dn + off;       // panel rows gi..gi+15
    const float* Arow1 = L + (size_t)(gi + 16 + m) * Dn + off;  // panel rows gi+16..gi+31
    const float* Brow0 = L + (size_t)(gj + m) * Dn + off;       // panel rows gj.. (used as B^T)
    const float* Brow1 = L + (size_t)(gj + 16 + m) * Dn + off;

#pragma unroll
    for (int k0 = 0; k0 < NB; k0 += 4) {
        v2f a0, a1, b0, b1;
        // negate A at staging: f32 WMMA NEG bits only allow C-negate, so fold the
        // subtraction (A22 -= L21*L21^T) into the A operand.
        a0.x = -Arow0[k0 + klo];  a0.y = -Arow0[k0 + klo + 1];
        a1.x = -Arow1[k0 + klo];  a1.y = -Arow1[k0 + klo + 1];
        b0.x =  Brow0[k0 + klo];  b0.y =  Brow0[k0 + klo + 1];
        b1.x =  Brow1[k0 + klo];  b1.y =  Brow1[k0 + klo + 1];
        c00 = __builtin_amdgcn_wmma_f32_16x16x4_f32(false, a0, false, b0, (short)0, c00, false, false);
        c01 = __builtin_amdgcn_wmma_f32_16x16x4_f32(false, a0, false, b1, (short)0, c01, false, false);
        c10 = __builtin_amdgcn_wmma_f32_16x16x4_f32(false, a1, false, b0, (short)0, c10, false, false);
        c11 = __builtin_amdgcn_wmma_f32_16x16x4_f32(false, a1, false, b1, (short)0, c11, false, false);
    }

#pragma unroll
    for (int r = 0; r < 8; ++r) {
        size_t row0 = (size_t)(gi + r + hi8) * Dn;
        size_t row1 = (size_t)(gi + 16 + r + hi8) * Dn;
        L[row0 + gj + m]      = c00[r];
        L[row0 + gj + 16 + m] = c01[r];
        L[row1 + gj + m]      = c10[r];
        L[row1 + gj + 16 + m] = c11[r];
    }
}

// ---------------- forward solve L z = (noisy - mu), plus z.z and sum(log diag) ----------------
__global__ void trsv_logdet(const float* __restrict__ noisy,
                            const float* __restrict__ muw,
                            const float* __restrict__ L,
                            float* __restrict__ scal) {
    __shared__ float zsh[Dn];
    __shared__ float accsh[Dn];
    __shared__ float s[NB][NB + 1];
    __shared__ float red[256];
    int tid = threadIdx.x; // 256 threads

    for (int i = tid; i < Dn; i += 256) accsh[i] = noisy[i] - muw[i];
    __syncthreads();

    for (int b = 0; b < Dn / NB; ++b) {
        int rb = b * NB;
        // stage diagonal block
        for (int idx = tid; idx < NB * NB; idx += 256) {
            int r = idx >> 6, cc = idx & 63;
            s[r][cc] = L[(size_t)(rb + r) * Dn + rb + cc];
        }
        __syncthreads();
        // wave-0 serial in-block solve (64 rows, 2 per lane, shfl broadcast, no barriers)
        if (tid < 32) {
            int lane = tid;
            float a0 = accsh[rb + lane];
            float a1 = accsh[rb + 32 + lane];
            for (int cc = 0; cc < NB; ++cc) {
                float num = __shfl((cc < 32) ? a0 : a1, cc & 31, 32);
                float zc = num / s[cc][cc];
                if (lane == 0) zsh[rb + cc] = zc;
                if (lane > cc)      a0 -= s[lane][cc]      * zc;
                if (32 + lane > cc) a1 -= s[32 + lane][cc] * zc;
            }
        }
        __syncthreads();
        // rank-64 GEMV update of all remaining rows
        for (int r = rb + NB + tid; r < Dn; r += 256) {
            const float4* lr = (const float4*)(L + (size_t)r * Dn + rb);
            float sum = 0.0f;
#pragma unroll
            for (int q = 0; q < NB / 4; ++q) {
                float4 lv = lr[q];
                sum += lv.x * zsh[rb + 4 * q]     + lv.y * zsh[rb + 4 * q + 1]
                     + lv.z * zsh[rb + 4 * q + 2] + lv.w * zsh[rb + 4 * q + 3];
            }
            accsh[r] -= sum;
        }
        __syncthreads();
    }

    // zz and log-det reductions
    float zz = 0.0f, ld = 0.0f;
    for (int j = tid; j < Dn; j += 256) {
        float z = zsh[j];
        zz += z * z;
        ld += logf(L[(size_t)j * Dn + j]);
    }
    red[tid] = zz; __syncthreads();
    for (int o = 128; o > 0; o >>= 1) { if (tid < o) red[tid] += red[tid + o]; __syncthreads(); }
    if (tid == 0) scal[1] = red[0];
    __syncthreads();
    red[tid] = ld; __syncthreads();
    for (int o = 128; o > 0; o >>= 1) { if (tid < o) red[tid] += red[tid + o]; __syncthreads(); }
    if (tid == 0) scal[2] = red[0];
}

// ---------------- big streaming reduction: S = sum (y - mu)^2 ----------------
__global__ void lik_partial(const float* __restrict__ y, const float* __restrict__ mu,
                            float* __restrict__ pars, int n4) {
    int i = blockIdx.x * blockDim.x + threadIdx.x;
    int stride = gridDim.x * blockDim.x;
    const float4* y4 = (const float4*)y;
    const float4* m4 = (const float4*)mu;
    float s = 0.0f;
    for (; i < n4; i += stride) {
        float4 a = y4[i], b = m4[i];
        float d0 = a.x - b.x, d1 = a.y - b.y, d2 = a.z - b.z, d3 = a.w - b.w;
        s += d0 * d0 + d1 * d1 + d2 * d2 + d3 * d3;
    }
    __shared__ float red[256];
    red[threadIdx.x] = s; __syncthreads();
    for (int o = 128; o > 0; o >>= 1) { if (threadIdx.x < o) red[threadIdx.x] += red[threadIdx.x + o]; __syncthreads(); }
    if (threadIdx.x == 0) pars[blockIdx.x] = red[0];
}

__global__ void reduce2(const float* __restrict__ pars, const float* __restrict__ noisy,
                        float* __restrict__ scal) {
    __shared__ float red[1024];
    int t = threadIdx.x;
    red[t] = pars[t]; __syncthreads();
    for (int o = 512; o > 0; o >>= 1) { if (t < o) red[t] += red[t + o]; __syncthreads(); }
    if (t == 0) scal[3] = red[0];
    __syncthreads();
    red[t] = noisy[t] * noisy[t] + noisy[t + 1024] * noisy[t + 1024];
    __syncthreads();
    for (int o = 512; o > 0; o >>= 1) { if (t < o) red[t] += red[t + o]; __syncthreads(); }
    if (t == 0) scal[0] = red[0];
}

__global__ void finalize_loss(const float* __restrict__ scal, const float* __restrict__ sigp,
                              float* __restrict__ out) {
    float P = scal[0], zz = scal[1], ld = scal[2], S = scal[3];
    float sig = sigp[0];
    const float Nf = 16777216.0f;
    float loss_prior = -0.5f * P - 0.5f * (float)Dn * LOG2PI;       // prior_sigma = 1
    float loss_lik   = -0.5f * S / (sig * sig) - Nf * (logf(sig) + 0.5f * LOG2PI);
    float loss_var   = -0.5f * zz - ld - 0.5f * (float)Dn * LOG2PI;
    out[0] = (loss_var - loss_prior) / Nf - loss_lik;
}

extern "C" void kernel_launch(void* const* d_in, const int* in_sizes, int n_in,
                              void* d_out, int out_size, void* d_ws, size_t ws_size,
                              hipStream_t stream) {
    const float* noisy = (const float*)d_in[0];
    const float* muw   = (const float*)d_in[1];
    const float* sigm  = (const float*)d_in[2];
    const float* mup   = (const float*)d_in[3];
    const float* sigp  = (const float*)d_in[4];
    const float* ytrue = (const float*)d_in[5];
    float* out = (float*)d_out;

    float* L    = (float*)d_ws;                 // 2048*2048 f32 = 16 MB
    float* pars = L + (size_t)Dn * Dn;          // 1024 partials
    float* scal = pars + 1024;                  // [P, zz, logdet, S]

    // independent streaming work first
    copy_mat<<<2048, 256, 0, stream>>>(sigm, L);
    lik_partial<<<1024, 256, 0, stream>>>(ytrue, mup, pars, 16777216 / 4);
    reduce2<<<1, 1024, 0, stream>>>(pars, noisy, scal);

    // blocked Cholesky, NB = 64, 32 steps
    for (int kb = 0; kb < Dn / NB; ++kb) {
        int off = kb * NB;
        chol_diag<<<1, NB, 0, stream>>>(L, off);
        int below = Dn - off - NB;
        if (below > 0) {
            trsm_panel<<<below / NB, NB, 0, stream>>>(L, off);
            int ntM = below / 32;                   // 32x32 macro-tiles
            int ntri = ntM * (ntM + 1) / 2;
            syrk_wmma<<<(ntri + 7) / 8, 256, 0, stream>>>(L, off, ntri);
        }
    }

    trsv_logdet<<<1, 256, 0, stream>>>(noisy, muw, L, scal);
    finalize_loss<<<1, 1, 0, stream>>>(scal, sigp, out);
}